// GIN_48919677501954
// MI455X (gfx1250) — compile-verified
//
#include <hip/hip_runtime.h>
#include <hip/hip_bf16.h>

// ---------------------------------------------------------------------------
// GIN forward on MI455X (gfx1250, wave32).
//   - edge scatter-add: bandwidth/atomic bound (float4 gather + f32 atomics)
//   - MLP GEMMs: v_wmma_f32_16x16x4_f32, one 16x16 tile per wave, K=128
//   - BN stats fused into GEMM epilogue (shfl_xor + atomicAdd), finalize,
//     then apply+ReLU (+ global_add_pool on last layer)
// ---------------------------------------------------------------------------

typedef __attribute__((ext_vector_type(2))) float v2f;
typedef __attribute__((ext_vector_type(8))) float v8f;

#define HID    128
#define LDSW   132   // padded row stride (floats): breaks stride-128 bank conflicts,
                     // keeps 16B alignment (132*4 = 528 = 33*16)

// --------------------------- simple data movers ----------------------------

__global__ __launch_bounds__(256) void copy_f4_kernel(float* __restrict__ dst,
                                                      const float* __restrict__ src,
                                                      int n4) {
    int t = blockIdx.x * 256 + threadIdx.x;
    if (t < n4) ((float4*)dst)[t] = ((const float4*)src)[t];
}

__global__ __launch_bounds__(256) void zero_kernel(float* __restrict__ p, int n) {
    int t = blockIdx.x * 256 + threadIdx.x;
    if (t < n) p[t] = 0.0f;
}

// --------------------------- edge scatter-add ------------------------------
// h[dst[e]][:] += x[src[e]][:]   (h pre-initialized with x  => h = x + agg)
// one wave per edge: 32 lanes x float4 = 128 channels.

__global__ __launch_bounds__(256) void scatter_add_kernel(float* __restrict__ h,
                                                          const float* __restrict__ x,
                                                          const int* __restrict__ ei,
                                                          int n_edges) {
    int t = blockIdx.x * 256 + threadIdx.x;
    int e = t >> 5;
    if (e >= n_edges) return;
    int lane = t & 31;
    int s = ei[e];            // edge_index[0][e]
    int d = ei[n_edges + e];  // edge_index[1][e]
    float4 v = ((const float4*)(x + (size_t)s * HID))[lane];
    float* o = h + (size_t)d * HID + lane * 4;
    atomicAdd(o + 0, v.x);
    atomicAdd(o + 1, v.y);
    atomicAdd(o + 2, v.z);
    atomicAdd(o + 3, v.w);
}

// ------------------- fused MLP (WMMA) + BN-stat accumulate -----------------
// In:  H = x + agg   (this block's 16 rows)   Out: H = h2 (in place, safe:
// rows are read into LDS before any global write, blocks own disjoint rows).

__global__ __launch_bounds__(256)
void gin_mlp_wmma_kernel(float* __restrict__ H,
                         const float* __restrict__ W1, const float* __restrict__ b1,
                         const float* __restrict__ W2, const float* __restrict__ b2,
                         float* __restrict__ sums, float* __restrict__ sumsq,
                         int n_nodes) {
    __shared__ float lds_h [16 * LDSW];
    __shared__ float lds_h1[16 * LDSW];

    const int tid  = threadIdx.x;
    const int wave = tid >> 5;        // 0..7  -> N-tile (16 output channels each)
    const int lane = tid & 31;
    const int hi   = lane >> 4;       // half-wave: selects K pair / M half
    const int l16  = lane & 15;
    const int m0   = blockIdx.x * 16; // base node row
    const int col  = wave * 16 + l16; // output channel owned by this lane

    // Weights are L2-resident across all blocks; hint the prefetcher.
    __builtin_prefetch(&W1[(size_t)tid * 64], 0, 0);   // 256 * 64 floats covers 128x128
    __builtin_prefetch(&W2[(size_t)tid * 64], 0, 0);

    // Stage this block's 16x128 activation tile into padded LDS.
    {
        const float4* hsrc = (const float4*)H;
        #pragma unroll
        for (int j = 0; j < 2; ++j) {
            int idx4 = tid + j * 256;          // 0..511 float4 slots
            int row  = idx4 >> 5;              // 16 rows
            int c4   = idx4 & 31;              // 32 float4 per row
            int grow = m0 + row;
            float4 v = make_float4(0.f, 0.f, 0.f, 0.f);
            if (grow < n_nodes) v = hsrc[(size_t)grow * 32 + c4];
            *(float4*)&lds_h[row * LDSW + c4 * 4] = v;
        }
    }
    __syncthreads();

    // ---- GEMM1: h1 = relu(h @ W1 + b1), 32 WMMAs over K=128 ----
    v8f c1 = {};
    #pragma unroll 4
    for (int kk = 0; kk < 32; ++kk) {
        int k0 = kk * 4 + hi * 2;
        v2f a = *(const v2f*)&lds_h[l16 * LDSW + k0];   // A: row=l16, K pair
        v2f b;
        b.x = W1[(size_t)k0 * HID + col];               // B: K x N row-major
        b.y = W1[(size_t)(k0 + 1) * HID + col];
        c1 = __builtin_amdgcn_wmma_f32_16x16x4_f32(false, a, false, b,
                                                   (short)0, c1, false, false);
    }
    {
        float bias1 = b1[col];
        #pragma unroll
        for (int r = 0; r < 8; ++r) {
            float v = fmaxf(c1[r] + bias1, 0.0f);
            lds_h1[(r + hi * 8) * LDSW + col] = v;      // D: vgpr r -> M = r + 8*hi
        }
    }
    __syncthreads();

    // ---- GEMM2: h2 = h1 @ W2 + b2 ----
    v8f c2 = {};
    #pragma unroll 4
    for (int kk = 0; kk < 32; ++kk) {
        int k0 = kk * 4 + hi * 2;
        v2f a = *(const v2f*)&lds_h1[l16 * LDSW + k0];
        v2f b;
        b.x = W2[(size_t)k0 * HID + col];
        b.y = W2[(size_t)(k0 + 1) * HID + col];
        c2 = __builtin_amdgcn_wmma_f32_16x16x4_f32(false, a, false, b,
                                                   (short)0, c2, false, false);
    }

    // Epilogue: store h2, accumulate per-channel sum / sum-of-squares.
    float bias2 = b2[col];
    float lsum = 0.0f, lsq = 0.0f;
    #pragma unroll
    for (int r = 0; r < 8; ++r) {
        int grow = m0 + r + hi * 8;
        float v = c2[r] + bias2;
        if (grow < n_nodes) {
            H[(size_t)grow * HID + col] = v;
            lsum += v;
            lsq  += v * v;
        }
    }
    // lanes l and l^16 hold the same channel (M halves) -> combine then atomic.
    lsum += __shfl_xor(lsum, 16, 32);
    lsq  += __shfl_xor(lsq, 16, 32);
    if (hi == 0) {
        atomicAdd(&sums[col],  lsum);
        atomicAdd(&sumsq[col], lsq);
    }
}

// ------------------------ BN finalize: scale/shift -------------------------

__global__ void bn_finalize_kernel(float* __restrict__ sums, float* __restrict__ sumsq,
                                   const float* __restrict__ gamma,
                                   const float* __restrict__ beta, float inv_n) {
    int c = threadIdx.x;  // 128
    float mean  = sums[c] * inv_n;
    float var   = sumsq[c] * inv_n - mean * mean;
    float scale = gamma[c] * rsqrtf(var + 1e-5f);
    sums[c]  = scale;                  // reuse buffers: sums -> scale
    sumsq[c] = beta[c] - mean * scale; // sumsq -> shift
}

// ----------------- BN apply + ReLU (+ global_add_pool last) ----------------

__global__ __launch_bounds__(256)
void bn_relu_pool_kernel(float* __restrict__ H, const float* __restrict__ scale,
                         const float* __restrict__ shift, float* __restrict__ pooled,
                         const int* __restrict__ batch, int n_nodes, int do_pool) {
    int t = blockIdx.x * 256 + threadIdx.x;
    int i  = t >> 5;
    int c4 = (t & 31) * 4;
    if (i >= n_nodes) return;
    float4 v  = *(float4*)&H[(size_t)i * HID + c4];
    float4 sc = *(const float4*)&scale[c4];
    float4 sh = *(const float4*)&shift[c4];
    v.x = fmaxf(v.x * sc.x + sh.x, 0.0f);
    v.y = fmaxf(v.y * sc.y + sh.y, 0.0f);
    v.z = fmaxf(v.z * sc.z + sh.z, 0.0f);
    v.w = fmaxf(v.w * sc.w + sh.w, 0.0f);
    *(float4*)&H[(size_t)i * HID + c4] = v;
    if (do_pool) {
        int g = batch[i];
        float* p = &pooled[(size_t)g * HID + c4];
        atomicAdd(p + 0, v.x);
        atomicAdd(p + 1, v.y);
        atomicAdd(p + 2, v.z);
        atomicAdd(p + 3, v.w);
    }
}

// ------------------------------ final linear -------------------------------

__global__ __launch_bounds__(256)
void final_linear_kernel(const float* __restrict__ pooled, const float* __restrict__ Wlin,
                         const float* __restrict__ blin, float* __restrict__ out,
                         int n_graphs, int n_out) {
    int t = blockIdx.x * 256 + threadIdx.x;
    if (t >= n_graphs * n_out) return;
    int g = t / n_out, o = t % n_out;
    float acc = blin[o];
    #pragma unroll 8
    for (int c = 0; c < HID; ++c)
        acc += pooled[(size_t)g * HID + c] * Wlin[(size_t)c * n_out + o];
    out[t] = acc;
}

// ------------------------------- launcher ----------------------------------

extern "C" void kernel_launch(void* const* d_in, const int* in_sizes, int n_in,
                              void* d_out, int out_size, void* d_ws, size_t ws_size,
                              hipStream_t stream) {
    const float* x      = (const float*)d_in[0];
    const float* W1s    = (const float*)d_in[1];
    const float* b1s    = (const float*)d_in[2];
    const float* W2s    = (const float*)d_in[3];
    const float* b2s    = (const float*)d_in[4];
    const float* gammas = (const float*)d_in[5];
    const float* betas  = (const float*)d_in[6];
    const float* Wlin   = (const float*)d_in[7];
    const float* blin   = (const float*)d_in[8];
    const int*   ei     = (const int*)d_in[9];
    const int*   batch  = (const int*)d_in[10];
    float*       out    = (float*)d_out;

    const int n_nodes  = in_sizes[0] / HID;     // 50000
    const int n_edges  = in_sizes[9] / 2;       // 800000
    const int n_graphs = 64;
    const int n_out    = in_sizes[8];           // 10

    // workspace: two ping-pong activation buffers + stats + pooled (contiguous)
    float* wsf    = (float*)d_ws;
    size_t nbuf   = (size_t)n_nodes * HID;
    float* bufA   = wsf;
    float* bufB   = wsf + nbuf;
    float* sums   = wsf + 2 * nbuf;             // 128
    float* sumsq  = sums + HID;                 // 128
    float* pooled = sumsq + HID;                // 64*128
    const int stats_n = HID + HID + n_graphs * HID;   // zeroed together

    dim3 blk(256);
    const int n4          = n_nodes * (HID / 4);
    const int copy_blocks = (n4 + 255) / 256;
    const int node_blocks = (n_nodes * 32 + 255) / 256;  // 32 threads per node
    const int scat_blocks = (n_edges * 32 + 255) / 256;  // 1 wave per edge
    const int mlp_blocks  = (n_nodes + 15) / 16;
    const int zero_blocks = (stats_n + 255) / 256;

    const float* x_cur = x;
    for (int l = 0; l < 3; ++l) {
        float* H = (l & 1) ? bufB : bufA;
        copy_f4_kernel<<<copy_blocks, blk, 0, stream>>>(H, x_cur, n4);
        scatter_add_kernel<<<scat_blocks, blk, 0, stream>>>(H, x_cur, ei, n_edges);
        zero_kernel<<<zero_blocks, blk, 0, stream>>>(sums, stats_n);
        gin_mlp_wmma_kernel<<<mlp_blocks, blk, 0, stream>>>(
            H, W1s + (size_t)l * HID * HID, b1s + (size_t)l * HID,
            W2s + (size_t)l * HID * HID, b2s + (size_t)l * HID,
            sums, sumsq, n_nodes);
        bn_finalize_kernel<<<1, HID, 0, stream>>>(sums, sumsq,
            gammas + (size_t)l * HID, betas + (size_t)l * HID, 1.0f / (float)n_nodes);
        bn_relu_pool_kernel<<<node_blocks, blk, 0, stream>>>(
            H, sums, sumsq, pooled, batch, n_nodes, (l == 2) ? 1 : 0);
        x_cur = H;
    }
    final_linear_kernel<<<(n_graphs * n_out + 255) / 256, blk, 0, stream>>>(
        pooled, Wlin, blin, out, n_graphs, n_out);
}